// GNBlock_90512140796488
// MI455X (gfx1250) — compile-verified
//
#include <hip/hip_runtime.h>
#include <hip/hip_bf16.h>

typedef __attribute__((ext_vector_type(2))) float v2f;
typedef __attribute__((ext_vector_type(8))) float v8f;

#define BATCH 16
#define N_V   1024
#define N_E   4096

// D = A(16x4) * B(4x16) + C(16x16), all fp32, wave32 WMMA.
__device__ __forceinline__ v8f wmma_f32_16x16x4(v2f a, v2f b, v8f c) {
    return __builtin_amdgcn_wmma_f32_16x16x4_f32(false, a, false, b, (short)0, c,
                                                 false, false);
}

// MLP tower on a 16-row tile staged in LDS x[16][68] (features in cols 0..63):
// 64 -> 32 (relu) -> 3x [32 -> 32 relu] -> 32 (linear), then mask + store.
// Weights row-major: W0[64][32], Wh[3][32][32], Wl[32][32].
// Output written to outBase[oa * ncols + n0 + el] (oa = 0..31), masked per column.
__device__ __forceinline__ void mlp16_64_32(
    float (*x)[68], int lane,
    const float* __restrict__ W0, const float* __restrict__ b0,
    const float* __restrict__ Wh, const float* __restrict__ bh,
    const float* __restrict__ Wl, const float* __restrict__ bl,
    const unsigned char* __restrict__ mask,   // per-batch mask base
    float* __restrict__ outBase,              // out + b*32*ncols
    int ncols, int n0)
{
    const int m  = lane & 15;
    const int kh = lane >> 4;

    // ---- layer 0: K = 64, two N-tiles of 16 ----
    v8f h0 = {}; v8f h1 = {};
    for (int k0 = 0; k0 < 64; k0 += 4) {
        int ka = k0 + 2 * kh;
        v2f a;  a.x  = x[m][ka];                a.y  = x[m][ka + 1];
        v2f w0; w0.x = W0[ka * 32 + m];         w0.y = W0[(ka + 1) * 32 + m];
        v2f w1; w1.x = W0[ka * 32 + 16 + m];    w1.y = W0[(ka + 1) * 32 + 16 + m];
        h0 = wmma_f32_16x16x4(a, w0, h0);
        h1 = wmma_f32_16x16x4(a, w1, h1);
    }
    {
        float bia0 = b0[m], bia1 = b0[16 + m];
        for (int j = 0; j < 8; ++j) {
            int e = j + 8 * kh;                 // row (edge/node) this lane owns
            x[e][m]      = fmaxf(h0[j] + bia0, 0.f);
            x[e][16 + m] = fmaxf(h1[j] + bia1, 0.f);
        }
    }

    // ---- hidden layers: K = 32 ----
    for (int l = 0; l < 3; ++l) {
        const float* W = Wh + l * 32 * 32;
        v8f g0 = {}; v8f g1 = {};
        for (int k0 = 0; k0 < 32; k0 += 4) {
            int ka = k0 + 2 * kh;
            v2f a;  a.x  = x[m][ka];             a.y  = x[m][ka + 1];
            v2f w0; w0.x = W[ka * 32 + m];       w0.y = W[(ka + 1) * 32 + m];
            v2f w1; w1.x = W[ka * 32 + 16 + m];  w1.y = W[(ka + 1) * 32 + 16 + m];
            g0 = wmma_f32_16x16x4(a, w0, g0);
            g1 = wmma_f32_16x16x4(a, w1, g1);
        }
        float bia0 = bh[l * 32 + m], bia1 = bh[l * 32 + 16 + m];
        for (int j = 0; j < 8; ++j) {
            int e = j + 8 * kh;
            x[e][m]      = fmaxf(g0[j] + bia0, 0.f);
            x[e][16 + m] = fmaxf(g1[j] + bia1, 0.f);
        }
    }

    // ---- final layer: K = 32, linear ----
    v8f o0 = {}; v8f o1 = {};
    for (int k0 = 0; k0 < 32; k0 += 4) {
        int ka = k0 + 2 * kh;
        v2f a;  a.x  = x[m][ka];                 a.y  = x[m][ka + 1];
        v2f w0; w0.x = Wl[ka * 32 + m];          w0.y = Wl[(ka + 1) * 32 + m];
        v2f w1; w1.x = Wl[ka * 32 + 16 + m];     w1.y = Wl[(ka + 1) * 32 + 16 + m];
        o0 = wmma_f32_16x16x4(a, w0, o0);
        o1 = wmma_f32_16x16x4(a, w1, o1);
    }
    {
        float bia0 = bl[m], bia1 = bl[16 + m];
        for (int j = 0; j < 8; ++j) {
            int e = j + 8 * kh;
            x[e][m]      = o0[j] + bia0;
            x[e][16 + m] = o1[j] + bia1;
        }
    }

    // ---- masked, coalesced transpose-store: out[oa][n0+el] ----
    for (int i = lane; i < 512; i += 32) {
        int oa = i >> 4, el = i & 15;
        float v = mask[n0 + el] ? x[el][oa] : 0.f;
        outBase[(size_t)oa * ncols + n0 + el] = v;
    }
}

// Edge pipeline: T_s = Va@Es, T_r = Va@Er (per 16-edge tile), features, MLP, mask.
__global__ __launch_bounds__(256)
void gn_edge_kernel(const float* __restrict__ Ea, const float* __restrict__ Es,
                    const float* __restrict__ Er, const float* __restrict__ Va,
                    const float* __restrict__ u,  const unsigned char* __restrict__ mask_e,
                    const float* __restrict__ W0, const float* __restrict__ b0,
                    const float* __restrict__ Wh, const float* __restrict__ bh,
                    const float* __restrict__ Wl, const float* __restrict__ bl,
                    float* __restrict__ outE)
{
    __shared__ float xs[8][16][68];
    const int lane = threadIdx.x & 31;
    const int wv   = threadIdx.x >> 5;
    const int gw   = blockIdx.x * 8 + wv;
    const int b    = gw >> 8;                 // 256 tiles per batch
    const int n0   = (gw & 255) << 4;
    const int m    = lane & 15;
    const int kh   = lane >> 4;

    const float* VaB = Va + (size_t)b * 16 * N_V;
    const float* EsB = Es + (size_t)b * N_V * N_E + n0 + m;
    const float* ErB = Er + (size_t)b * N_V * N_E + n0 + m;

    v8f accS = {}; v8f accR = {};
    for (int k0 = 0; k0 < N_V; k0 += 4) {
        int ka = k0 + 2 * kh;
        v2f a;  a.x  = VaB[m * N_V + ka];        a.y  = VaB[m * N_V + ka + 1];
        v2f bs; bs.x = EsB[(size_t)ka * N_E];    bs.y = EsB[(size_t)(ka + 1) * N_E];
        v2f br; br.x = ErB[(size_t)ka * N_E];    br.y = ErB[(size_t)(ka + 1) * N_E];
        accS = wmma_f32_16x16x4(a, bs, accS);    // one A fragment feeds both GEMMs
        accR = wmma_f32_16x16x4(a, br, accR);
    }

    float (*x)[68] = xs[wv];
    for (int i = 0; i < 8; ++i) {                // E_a -> feats 0..15, u -> 48..63
        int f = kh * 8 + i;
        x[m][f]      = Ea[((size_t)b * 16 + f) * N_E + n0 + m];
        x[m][48 + f] = u[b * 16 + f];
    }
    for (int j = 0; j < 8; ++j) {                // T_s -> 16..31, T_r -> 32..47
        x[m][16 + kh * 8 + j] = accS[j];
        x[m][32 + kh * 8 + j] = accR[j];
    }

    mlp16_64_32(x, lane, W0, b0, Wh, bh, Wl, bl,
                mask_e + (size_t)b * N_E, outE + (size_t)b * 32 * N_E, N_E, n0);
}

// Node pipeline: M = outE @ Er^T (per 16-node tile), features, MLP, mask.
__global__ __launch_bounds__(256)
void gn_node_kernel(const float* __restrict__ Va, const float* __restrict__ Er,
                    const float* __restrict__ outE, const float* __restrict__ u,
                    const unsigned char* __restrict__ mask_v,
                    const float* __restrict__ W0, const float* __restrict__ b0,
                    const float* __restrict__ Wh, const float* __restrict__ bh,
                    const float* __restrict__ Wl, const float* __restrict__ bl,
                    float* __restrict__ outV)
{
    __shared__ float xs[8][16][68];
    const int lane = threadIdx.x & 31;
    const int wv   = threadIdx.x >> 5;
    const int gw   = blockIdx.x * 8 + wv;
    const int b    = gw >> 6;                 // 64 tiles per batch
    const int n0   = (gw & 63) << 4;
    const int m    = lane & 15;
    const int kh   = lane >> 4;

    const float* A   = outE + (size_t)b * 32 * N_E;                   // [32, N_E]
    const float* ErB = Er + ((size_t)b * N_V + (n0 + m)) * N_E;       // node row

    v8f acc0 = {}; v8f acc1 = {};
    for (int k0 = 0; k0 < N_E; k0 += 4) {
        int ka = k0 + 2 * kh;
        v2f a0; a0.x = A[(size_t)m * N_E + ka];         a0.y = A[(size_t)m * N_E + ka + 1];
        v2f a1; a1.x = A[(size_t)(16 + m) * N_E + ka];  a1.y = A[(size_t)(16 + m) * N_E + ka + 1];
        v2f bf; bf.x = ErB[ka];                         bf.y = ErB[ka + 1];
        acc0 = wmma_f32_16x16x4(a0, bf, acc0);          // rows 0..15 of M
        acc1 = wmma_f32_16x16x4(a1, bf, acc1);          // rows 16..31 of M
    }

    float (*x)[68] = xs[wv];
    for (int i = 0; i < 8; ++i) {                // V_a -> feats 0..15, u -> 48..63
        int f = kh * 8 + i;
        x[m][f]      = Va[((size_t)b * 16 + f) * N_V + n0 + m];
        x[m][48 + f] = u[b * 16 + f];
    }
    for (int j = 0; j < 8; ++j) {                // M rows 0..15 -> 16..31, 16..31 -> 32..47
        x[m][16 + kh * 8 + j] = acc0[j];
        x[m][32 + kh * 8 + j] = acc1[j];
    }

    mlp16_64_32(x, lane, W0, b0, Wh, bh, Wl, bl,
                mask_v + (size_t)b * N_V, outV + (size_t)b * 32 * N_V, N_V, n0);
}

// Global pipeline: sum-pool outE/outV per batch, then tiny 80->...->16 MLP.
__global__ __launch_bounds__(256)
void gn_global_kernel(const float* __restrict__ u, const float* __restrict__ outE,
                      const float* __restrict__ outV,
                      const float* __restrict__ W0, const float* __restrict__ b0,
                      const float* __restrict__ Wh, const float* __restrict__ bh,
                      const float* __restrict__ Wl, const float* __restrict__ bl,
                      float* __restrict__ outU)
{
    __shared__ float P[80];
    __shared__ float redV[8], redE[8];
    __shared__ float h[32], h2[32];
    const int b = blockIdx.x;
    const int tid = threadIdx.x;

    if (tid < 16) P[tid] = u[b * 16 + tid];

    for (int r = 0; r < 32; ++r) {
        const float* rowV = outV + ((size_t)b * 32 + r) * N_V;
        const float* rowE = outE + ((size_t)b * 32 + r) * N_E;
        float sv = 0.f, se = 0.f;
        for (int i = tid; i < N_V; i += 256) sv += rowV[i];
        for (int i = tid; i < N_E; i += 256) se += rowE[i];
        for (int off = 16; off > 0; off >>= 1) {
            sv += __shfl_down(sv, off, 32);
            se += __shfl_down(se, off, 32);
        }
        if ((tid & 31) == 0) { redV[tid >> 5] = sv; redE[tid >> 5] = se; }
        __syncthreads();
        if (tid == 0) {
            float tv = 0.f, te = 0.f;
            for (int w = 0; w < 8; ++w) { tv += redV[w]; te += redE[w]; }
            P[16 + r] = tv; P[48 + r] = te;
        }
        __syncthreads();
    }

    if (tid < 32) {
        float s = b0[tid];
        for (int k = 0; k < 80; ++k) s += P[k] * W0[k * 32 + tid];
        h[tid] = fmaxf(s, 0.f);
    }
    __syncthreads();
    for (int l = 0; l < 3; ++l) {
        if (tid < 32) {
            float s = bh[l * 32 + tid];
            for (int k = 0; k < 32; ++k) s += h[k] * Wh[(l * 32 + k) * 32 + tid];
            h2[tid] = fmaxf(s, 0.f);
        }
        __syncthreads();
        if (tid < 32) h[tid] = h2[tid];
        __syncthreads();
    }
    if (tid < 16) {
        float s = bl[tid];
        for (int k = 0; k < 32; ++k) s += h[k] * Wl[k * 16 + tid];
        outU[b * 16 + tid] = s;
    }
}

extern "C" void kernel_launch(void* const* d_in, const int* in_sizes, int n_in,
                              void* d_out, int out_size, void* d_ws, size_t ws_size,
                              hipStream_t stream) {
    (void)in_sizes; (void)n_in; (void)out_size; (void)d_ws; (void)ws_size;

    const float* Ea = (const float*)d_in[0];
    const float* Es = (const float*)d_in[1];
    const float* Er = (const float*)d_in[2];
    const float* Va = (const float*)d_in[3];
    const float* u  = (const float*)d_in[4];
    const unsigned char* mask_v = (const unsigned char*)d_in[5];
    const unsigned char* mask_e = (const unsigned char*)d_in[6];
    const float* feW0 = (const float*)d_in[7];
    const float* feb0 = (const float*)d_in[8];
    const float* feWh = (const float*)d_in[9];
    const float* febh = (const float*)d_in[10];
    const float* feWl = (const float*)d_in[11];
    const float* febl = (const float*)d_in[12];
    const float* fvW0 = (const float*)d_in[13];
    const float* fvb0 = (const float*)d_in[14];
    const float* fvWh = (const float*)d_in[15];
    const float* fvbh = (const float*)d_in[16];
    const float* fvWl = (const float*)d_in[17];
    const float* fvbl = (const float*)d_in[18];
    const float* fuW0 = (const float*)d_in[19];
    const float* fub0 = (const float*)d_in[20];
    const float* fuWh = (const float*)d_in[21];
    const float* fubh = (const float*)d_in[22];
    const float* fuWl = (const float*)d_in[23];
    const float* fubl = (const float*)d_in[24];

    float* out  = (float*)d_out;
    float* outE = out;                                       // [16, 32, 4096]
    float* outV = out + (size_t)BATCH * 32 * N_E;            // [16, 32, 1024]
    float* outU = outV + (size_t)BATCH * 32 * N_V;           // [16, 16, 1]

    // 16 batches * 256 edge tiles = 4096 waves / 8 waves-per-block
    gn_edge_kernel<<<512, 256, 0, stream>>>(Ea, Es, Er, Va, u, mask_e,
                                            feW0, feb0, feWh, febh, feWl, febl, outE);
    // 16 batches * 64 node tiles = 1024 waves / 8
    gn_node_kernel<<<128, 256, 0, stream>>>(Va, Er, outE, u, mask_v,
                                            fvW0, fvb0, fvWh, fvbh, fvWl, fvbl, outV);
    // one block per batch
    gn_global_kernel<<<16, 256, 0, stream>>>(u, outE, outV,
                                             fuW0, fub0, fuWh, fubh, fuWl, fubl, outU);
}